// SMPL_40389872451788
// MI455X (gfx1250) — compile-verified
//
#include <hip/hip_runtime.h>
#include <hip/hip_bf16.h>
#include <math.h>

typedef __attribute__((ext_vector_type(2))) float v2f;
typedef __attribute__((ext_vector_type(8))) float v8f;

#define B_    512
#define V_    6890
#define J_    24
#define NB_   10
#define P_    207      // pose-feature length (K of big GEMM)
#define PK_   208      // K padded to multiple of 4
#define N3_   20670    // V*3 (N of big GEMM)
#define NPAD_ 20672    // N padded to multiple of 16
#define NTILES_ (NPAD_/16)   // 1292
#define KSTEPS_ (PK_/4)      // 52

__constant__ int c_parents[J_] = {-1,0,0,0,1,2,3,4,5,6,7,8,9,9,9,12,13,14,16,17,18,19,20,21};

// ---------------------------------------------------------------------------
// K0: zero-pad posedirs (207 x 20670) -> (208 x 20672)
// ---------------------------------------------------------------------------
__global__ void k_pad_posedirs(const float* __restrict__ pd, float* __restrict__ pdp) {
  int c = blockIdx.x * 256 + threadIdx.x;
  int r = blockIdx.y;
  if (c >= NPAD_) return;
  float v = (r < P_ && c < N3_) ? pd[(size_t)r * N3_ + c] : 0.0f;
  pdp[(size_t)r * NPAD_ + c] = v;
}

// ---------------------------------------------------------------------------
// K1: batch-independent constants C[j][0..2] = Jreg@v_template,
//     C[j][3 + k*10 + l] = Jreg@shapedirs   (24 x 33 floats)
// ---------------------------------------------------------------------------
__global__ void k_consts(const float* __restrict__ Jreg, const float* __restrict__ vt,
                         const float* __restrict__ sd, float* __restrict__ C) {
  __shared__ float red[256];
  int j   = blockIdx.x / 33;
  int idx = blockIdx.x % 33;
  float acc = 0.0f;
  for (int v = threadIdx.x; v < V_; v += 256) {
    float w = Jreg[(size_t)j * V_ + v];
    float x = (idx < 3) ? vt[v * 3 + idx] : sd[(size_t)v * 30 + (idx - 3)];
    acc += w * x;
  }
  red[threadIdx.x] = acc;
  __syncthreads();
  for (int s = 128; s > 0; s >>= 1) {
    if (threadIdx.x < s) red[threadIdx.x] += red[threadIdx.x + s];
    __syncthreads();
  }
  if (threadIdx.x == 0) C[j * 33 + idx] = red[0];
}

// ---------------------------------------------------------------------------
// K2: v_shaped[b,v,k] = v_template + shapedirs @ betas[b]
// ---------------------------------------------------------------------------
__global__ void k_shape(const float* __restrict__ betas, const float* __restrict__ vt,
                        const float* __restrict__ sd, float* __restrict__ vs) {
  int b = blockIdx.y;
  __shared__ float sb[NB_];
  if (threadIdx.x < NB_) sb[threadIdx.x] = betas[b * NB_ + threadIdx.x];
  __syncthreads();
  int v = blockIdx.x * 256 + threadIdx.x;
  if (v >= V_) return;
  const float* s = sd + (size_t)v * 30;
  float* o = vs + ((size_t)b * V_ + v) * 3;
#pragma unroll
  for (int k = 0; k < 3; k++) {
    float a = vt[v * 3 + k];
#pragma unroll
    for (int l = 0; l < NB_; l++) a += sb[l] * s[k * 10 + l];
    o[k] = a;
  }
}

// ---------------------------------------------------------------------------
// K3: per-batch joint pipeline: Rodrigues -> pose_feature (padded row of 208),
//     Jpos (from constants), kinematic chain, joints out, A matrices (24 x 12)
//     One wave (32 threads) per batch element; lane = joint.
// ---------------------------------------------------------------------------
__global__ void __launch_bounds__(32) k_pose(
    const float* __restrict__ go, const float* __restrict__ bp,
    const float* __restrict__ betas, const float* __restrict__ transl,
    const float* __restrict__ C, float* __restrict__ PF,
    float* __restrict__ Amat, float* __restrict__ outJoints) {
  int b = blockIdx.x;
  int j = threadIdx.x;
  __shared__ float sT[J_][16];
  __shared__ float sJ[J_][3];
  float R[9];
  float jp[3] = {0.f, 0.f, 0.f};

  if (j < J_) {
    float ax, ay, az;
    if (j == 0) { ax = go[b*3]; ay = go[b*3+1]; az = go[b*3+2]; }
    else { const float* p = bp + (size_t)b*69 + (j-1)*3; ax = p[0]; ay = p[1]; az = p[2]; }
    float ang = sqrtf(ax*ax + ay*ay + az*az + 1e-8f);
    float inv = 1.0f / ang;
    float x = ax*inv, y = ay*inv, z = az*inv;
    float c = cosf(ang), s = sinf(ang), t = 1.0f - c;
    R[0]=c+t*x*x;   R[1]=t*x*y-s*z; R[2]=t*x*z+s*y;
    R[3]=t*x*y+s*z; R[4]=c+t*y*y;   R[5]=t*y*z-s*x;
    R[6]=t*x*z-s*y; R[7]=t*y*z+s*x; R[8]=c+t*z*z;

    if (j >= 1) {
      float* pf = PF + (size_t)b * PK_ + (j-1)*9;
#pragma unroll
      for (int i = 0; i < 9; i++)
        pf[i] = R[i] - ((i==0 || i==4 || i==8) ? 1.0f : 0.0f);
    } else {
      PF[(size_t)b * PK_ + 207] = 0.0f;   // K-pad column
    }

    float beta[NB_];
#pragma unroll
    for (int l = 0; l < NB_; l++) beta[l] = betas[b*NB_ + l];
#pragma unroll
    for (int k = 0; k < 3; k++) {
      float a = C[j*33 + k];
#pragma unroll
      for (int l = 0; l < NB_; l++) a += beta[l] * C[j*33 + 3 + k*10 + l];
      jp[k] = a; sJ[j][k] = a;
    }
  }
  __syncthreads();

  if (j < J_) {
    int p = c_parents[j];
    float rel[3];
#pragma unroll
    for (int k = 0; k < 3; k++) rel[k] = jp[k] - ((p >= 0) ? sJ[p][k] : 0.0f);
#pragma unroll
    for (int m = 0; m < 3; m++) {
#pragma unroll
      for (int n = 0; n < 3; n++) sT[j][m*4+n] = R[m*3+n];
      sT[j][m*4+3] = rel[m];
    }
    sT[j][12] = 0.f; sT[j][13] = 0.f; sT[j][14] = 0.f; sT[j][15] = 1.f;
  }
  __syncthreads();

  if (j < J_) {
    // walk root->j, multiplying 4x4 transforms (max depth 8)
    int path[10]; int d = 0; int p = j;
    while (p >= 0) { path[d++] = p; p = c_parents[p]; }
    float M[16];
    int r0 = path[d-1];
#pragma unroll
    for (int i = 0; i < 16; i++) M[i] = sT[r0][i];
    for (int q = d - 2; q >= 0; --q) {
      int ci = path[q];
      float T2[16];
#pragma unroll
      for (int i = 0; i < 16; i++) T2[i] = sT[ci][i];
      float Rm[16];
#pragma unroll
      for (int m = 0; m < 4; m++)
#pragma unroll
        for (int n = 0; n < 4; n++) {
          float a = 0.f;
#pragma unroll
          for (int kk = 0; kk < 4; kk++) a += M[m*4+kk] * T2[kk*4+n];
          Rm[m*4+n] = a;
        }
#pragma unroll
      for (int i = 0; i < 16; i++) M[i] = Rm[i];
    }
    // posed joints (+ transl) straight to output
    size_t joff = (size_t)(b * J_ + j) * 3;
#pragma unroll
    for (int k = 0; k < 3; k++) outJoints[joff + k] = M[k*4+3] + transl[b*3+k];
    // A = transforms with translation column corrected by -R*Jpos (rows 0..2)
    float* Ao = Amat + ((size_t)b * J_ + j) * 12;
#pragma unroll
    for (int m = 0; m < 3; m++) {
      float tj = M[m*4]*jp[0] + M[m*4+1]*jp[1] + M[m*4+2]*jp[2];
      Ao[m*4+0] = M[m*4+0]; Ao[m*4+1] = M[m*4+1];
      Ao[m*4+2] = M[m*4+2]; Ao[m*4+3] = M[m*4+3] - tj;
    }
  }
}

// ---------------------------------------------------------------------------
// K4: WMMA GEMM: pose_offsets = PF(512x208) @ PDpad(208x20672), fused
//     v_posed = v_shaped + pose_offsets written in place into VS.
//     4 waves/block; each wave: one 16-col strip x 8 M-tiles (128 rows),
//     so each strided B-operand load feeds 8 v_wmma_f32_16x16x4_f32.
// ---------------------------------------------------------------------------
__global__ void __launch_bounds__(128) k_gemm(const float* __restrict__ PF,
                                              const float* __restrict__ PD,
                                              float* __restrict__ VS) {
  int wave   = threadIdx.x >> 5;
  int lane   = threadIdx.x & 31;
  int lane16 = lane & 15;
  int half   = lane >> 4;
  int nTile  = blockIdx.x * 4 + wave;
  int mGroup = blockIdx.y;                 // 0..3 -> 128 rows each
  int col    = nTile * 16 + lane16;        // B/D column (N)
  int rowB   = mGroup * 128 + lane16;      // A row (M)

  const float* pdc = PD + (size_t)(2 * half) * NPAD_ + col;

  v8f zero = {0.f,0.f,0.f,0.f,0.f,0.f,0.f,0.f};
  v8f acc[8];
#pragma unroll
  for (int mt = 0; mt < 8; mt++) acc[mt] = zero;

  for (int k = 0; k < KSTEPS_; k++) {
    const float* pp = pdc + (size_t)k * 4 * NPAD_;
    v2f bm;
    bm.x = pp[0];          // K = 4k + 2*half
    bm.y = pp[NPAD_];      // K = 4k + 2*half + 1
    __builtin_prefetch(pp + (size_t)4 * NPAD_, 0, 1);
    int k4 = k * 4 + 2 * half;
#pragma unroll
    for (int mt = 0; mt < 8; mt++) {
      v2f a = *(const v2f*)(PF + (size_t)(rowB + mt * 16) * PK_ + k4);
      acc[mt] = __builtin_amdgcn_wmma_f32_16x16x4_f32(
          false, a, false, bm, (short)0, acc[mt], false, false);
    }
  }

  if (col < N3_) {
#pragma unroll
    for (int mt = 0; mt < 8; mt++) {
      int mrow = mGroup * 128 + mt * 16 + 8 * half;
#pragma unroll
      for (int r = 0; r < 8; r++) {
        size_t idx = (size_t)(mrow + r) * N3_ + col;   // (b, flat v*3+k)
        VS[idx] += acc[mt][r];                          // v_posed in place
      }
    }
  }
}

// ---------------------------------------------------------------------------
// K5: LBS: T = lbs_weights @ A (per batch, A staged in LDS), then
//     verts = T[:3,:3] @ v_posed + T[:3,3] + transl
// ---------------------------------------------------------------------------
__global__ void __launch_bounds__(256) k_lbs(const float* __restrict__ W,
                                             const float* __restrict__ Amat,
                                             const float* __restrict__ VP,
                                             const float* __restrict__ transl,
                                             float* __restrict__ outV) {
  int b = blockIdx.y;
  __shared__ float sA[J_ * 12];
  for (int i = threadIdx.x; i < J_ * 12; i += 256)
    sA[i] = Amat[(size_t)b * J_ * 12 + i];
  __syncthreads();
  int v = blockIdx.x * 256 + threadIdx.x;
  if (v >= V_) return;
  const float* wr = W + (size_t)v * J_;
  float T[12];
#pragma unroll
  for (int i = 0; i < 12; i++) T[i] = 0.f;
#pragma unroll
  for (int j = 0; j < J_; j++) {
    float w = wr[j];
#pragma unroll
    for (int i = 0; i < 12; i++) T[i] += w * sA[j * 12 + i];
  }
  const float* vp = VP + ((size_t)b * V_ + v) * 3;
  float px = vp[0], py = vp[1], pz = vp[2];
  float tx = transl[b*3], ty = transl[b*3+1], tz = transl[b*3+2];
  float* o = outV + ((size_t)b * V_ + v) * 3;
  o[0] = T[0]*px + T[1]*py + T[2]*pz  + T[3]  + tx;
  o[1] = T[4]*px + T[5]*py + T[6]*pz  + T[7]  + ty;
  o[2] = T[8]*px + T[9]*py + T[10]*pz + T[11] + tz;
}

// ---------------------------------------------------------------------------
extern "C" void kernel_launch(void* const* d_in, const int* in_sizes, int n_in,
                              void* d_out, int out_size, void* d_ws, size_t ws_size,
                              hipStream_t stream) {
  const float* betas  = (const float*)d_in[0];
  const float* go     = (const float*)d_in[1];
  const float* bp     = (const float*)d_in[2];
  const float* transl = (const float*)d_in[3];
  const float* vt     = (const float*)d_in[4];
  const float* sd     = (const float*)d_in[5];
  const float* pd     = (const float*)d_in[6];
  const float* Jreg   = (const float*)d_in[7];
  const float* W      = (const float*)d_in[8];

  float* ws = (float*)d_ws;
  size_t o = 0;
  float* PDpad = ws + o;  o += (size_t)PK_ * NPAD_;       // 4,299,776
  float* PFpad = ws + o;  o += (size_t)B_ * PK_;          //   106,496
  float* VS    = ws + o;  o += (size_t)B_ * V_ * 3;       // 10,583,040
  float* Amat  = ws + o;  o += (size_t)B_ * J_ * 12;      //   147,456
  float* C     = ws + o;  o += (size_t)J_ * 33;           //       792

  float* outVerts  = (float*)d_out;
  float* outJoints = outVerts + (size_t)B_ * V_ * 3;

  k_pad_posedirs<<<dim3((NPAD_ + 255) / 256, PK_), 256, 0, stream>>>(pd, PDpad);
  k_consts<<<dim3(J_ * 33), 256, 0, stream>>>(Jreg, vt, sd, C);
  k_shape<<<dim3((V_ + 255) / 256, B_), 256, 0, stream>>>(betas, vt, sd, VS);
  k_pose<<<dim3(B_), 32, 0, stream>>>(go, bp, betas, transl, C, PFpad, Amat, outJoints);
  k_gemm<<<dim3(NTILES_ / 4, B_ / 128), 128, 0, stream>>>(PFpad, PDpad, VS);
  k_lbs<<<dim3((V_ + 255) / 256, B_), 256, 0, stream>>>(W, Amat, VS, transl, outVerts);
}